// GCN_45509473469206
// MI455X (gfx1250) — compile-verified
//
#include <hip/hip_runtime.h>
#include <hip/hip_bf16.h>
#include <math.h>

typedef __attribute__((ext_vector_type(2))) float v2f;
typedef __attribute__((ext_vector_type(8))) float v8f;
typedef int v4i __attribute__((vector_size(16)));

#if __has_builtin(__builtin_amdgcn_global_load_async_to_lds_b128)
#define HAVE_ASYNC_LDS 1
#endif

__device__ __forceinline__ void wait_asynccnt0() {
#if __has_builtin(__builtin_amdgcn_s_wait_asynccnt)
    __builtin_amdgcn_s_wait_asynccnt(0);
#else
    asm volatile("s_wait_asynccnt 0" ::: "memory");
#endif
}

// ---------------------------------------------------------------- utilities
__global__ void k_fill(float* __restrict__ p, float v, int n) {
    int i = blockIdx.x * blockDim.x + threadIdx.x;
    if (i < n) p[i] = v;
}

__global__ void k_deg(const int* __restrict__ dstIdx, float* __restrict__ deg, int E) {
    int e = blockIdx.x * blockDim.x + threadIdx.x;
    if (e < E) unsafeAtomicAdd(&deg[dstIdx[e]], 1.0f);
}

__global__ void k_rsqrt(const float* __restrict__ deg, float* __restrict__ dis, int n) {
    int i = blockIdx.x * blockDim.x + threadIdx.x;
    if (i < n) dis[i] = rsqrtf(deg[i]);
}

__global__ void k_norm(const int* __restrict__ srcIdx, const int* __restrict__ dstIdx,
                       const float* __restrict__ dis, float* __restrict__ norm, int E) {
    int e = blockIdx.x * blockDim.x + threadIdx.x;
    if (e < E) norm[e] = dis[srcIdx[e]] * dis[dstIdx[e]];
}

// ------------------------------------------------------- GEMM1: h = x @ W1
// x:[n,256] row-major, W1:[256,128] row-major, h:[n,128].
// Block = 256 threads = 8 waves; block owns 16 rows x all 128 cols.
// x tile (16x256) staged in LDS (pad-stride 260 to kill bank conflicts) via
// GLOBAL_LOAD_ASYNC_TO_LDS_B128 when available (ASYNCcnt path), else sync.
// Wave w computes the 16x16 tile at columns [16w,16w+16) with 64 chained
// V_WMMA_F32_16X16X4_F32 ops (K loop over 256 in steps of 4).
__global__ void __launch_bounds__(256) k_gemm1(const float* __restrict__ x,
                                               const float* __restrict__ W1,
                                               float* __restrict__ h, int n) {
    __shared__ float xs[16 * 260];
    const int tid  = threadIdx.x;
    const int row0 = blockIdx.x * 16;
    const bool fullTile = (row0 + 16 <= n);   // uniform branch

    if (fullTile) {
#ifdef HAVE_ASYNC_LDS
        // async DMA: 1024 x b128, 4 per thread; bypasses VGPRs entirely
        for (int i = tid; i < 16 * 64; i += 256) {
            const int r = i >> 6;
            const int c = (i & 63) << 2;
            __builtin_amdgcn_global_load_async_to_lds_b128(
                (__attribute__((address_space(1))) v4i*)(x + (size_t)(row0 + r) * 256 + c),
                (__attribute__((address_space(3))) v4i*)&xs[r * 260 + c],
                0, 0);
        }
        wait_asynccnt0();
#else
        for (int i = tid; i < 16 * 64; i += 256) {
            const int r = i >> 6;
            const int c = (i & 63) << 2;
            *(float4*)&xs[r * 260 + c] =
                *(const float4*)(x + (size_t)(row0 + r) * 256 + c);
        }
#endif
    } else {
        for (int i = tid; i < 16 * 64; i += 256) {
            const int r = i >> 6;
            const int c = (i & 63) << 2;
            float4 v = make_float4(0.f, 0.f, 0.f, 0.f);
            if (row0 + r < n)
                v = *(const float4*)(x + (size_t)(row0 + r) * 256 + c);
            *(float4*)&xs[r * 260 + c] = v;
        }
    }
    __syncthreads();

    const int lane = tid & 31;
    const int wave = tid >> 5;
    const int half = lane >> 4;   // lanes 16-31 hold the K+2/K+3 slice
    const int mn   = lane & 15;   // A-row / B-col index within tile
    const int col0 = wave << 4;

    v8f acc = {0.f, 0.f, 0.f, 0.f, 0.f, 0.f, 0.f, 0.f};
    for (int k0 = 0; k0 < 256; k0 += 4) {
        const int ka = k0 + (half << 1);
        v2f a; a.x = xs[mn * 260 + ka];
               a.y = xs[mn * 260 + ka + 1];
        v2f b; b.x = W1[(size_t)ka       * 128 + col0 + mn];
               b.y = W1[(size_t)(ka + 1) * 128 + col0 + mn];
        acc = __builtin_amdgcn_wmma_f32_16x16x4_f32(
            /*neg_a=*/false, a, /*neg_b=*/false, b,
            /*c_mod=*/(short)0, acc, /*reuse_a=*/false, /*reuse_b=*/false);
    }

    // D layout: VGPR r -> M = r + 8*half (per-lane half), N = mn
    const int mBase = half << 3;
    if (fullTile) {
#pragma unroll
        for (int r = 0; r < 8; ++r)
            h[(size_t)(row0 + mBase + r) * 128 + col0 + mn] = acc[r];
    } else {
#pragma unroll
        for (int r = 0; r < 8; ++r) {
            const int row = row0 + mBase + r;
            if (row < n) h[(size_t)row * 128 + col0 + mn] = acc[r];
        }
    }
}

// ---------------------------------------------- edge scatter, layer 1 (H=128)
// One wave per edge: lane carries 4 contiguous channels (coalesced 512B read),
// 4 global fp32 atomics per lane into agg[dst]. This is the roofline-dominant
// kernel (~2 GB of random HBM traffic) — pure bandwidth, no flops to hide.
__global__ void __launch_bounds__(256) k_agg1(const int* __restrict__ srcIdx,
                                              const int* __restrict__ dstIdx,
                                              const float* __restrict__ norm,
                                              const float* __restrict__ h,
                                              float* __restrict__ agg, int E) {
    const int wave = (blockIdx.x * blockDim.x + threadIdx.x) >> 5;
    const int lane = threadIdx.x & 31;
    if (wave >= E) return;
    const int   s   = srcIdx[wave];
    const int   d   = dstIdx[wave];
    const float nrm = norm[wave];
    const float4 v = *(const float4*)(h + (size_t)s * 128 + lane * 4);
    float* ap = agg + (size_t)d * 128 + lane * 4;
    unsafeAtomicAdd(ap + 0, v.x * nrm);
    unsafeAtomicAdd(ap + 1, v.y * nrm);
    unsafeAtomicAdd(ap + 2, v.z * nrm);
    unsafeAtomicAdd(ap + 3, v.w * nrm);
}

// ------------------------- self-loop + bias + relu (in place: agg -> h1)
__global__ void k_bias_relu(const float* __restrict__ h, const float* __restrict__ dis,
                            const float* __restrict__ b1, float* agg, int n) {
    const int t = blockIdx.x * blockDim.x + threadIdx.x;   // n*32 threads
    if (t >= n * 32) return;
    const int i = t >> 5;
    const int c = (t & 31) << 2;
    const float d2 = dis[i] * dis[i];
    float4 a  = *(float4*)(agg + (size_t)i * 128 + c);
    float4 hh = *(const float4*)(h + (size_t)i * 128 + c);
    float4 bb = *(const float4*)(b1 + c);
    float4 r;
    r.x = fmaxf(fmaf(hh.x, d2, a.x) + bb.x, 0.f);
    r.y = fmaxf(fmaf(hh.y, d2, a.y) + bb.y, 0.f);
    r.z = fmaxf(fmaf(hh.z, d2, a.z) + bb.z, 0.f);
    r.w = fmaxf(fmaf(hh.w, d2, a.w) + bb.w, 0.f);
    *(float4*)(agg + (size_t)i * 128 + c) = r;
}

// ------------------------------------------------------- GEMM2: h2 = h1 @ W2
// h1:[n,128], W2:[128,16], h2:[n,16]. 8 waves/block, wave owns 16 rows x 16 cols,
// K loop of 32 WMMA f32 16x16x4 ops. W2 (8 KB) stays hot in WGP$/L2.
__global__ void __launch_bounds__(256) k_gemm2(const float* __restrict__ h1,
                                               const float* __restrict__ W2,
                                               float* __restrict__ h2, int n) {
    const int tid  = threadIdx.x;
    const int lane = tid & 31;
    const int wave = tid >> 5;
    const int half = lane >> 4;
    const int mn   = lane & 15;
    const int row0 = blockIdx.x * 128 + wave * 16;
    const int m    = row0 + mn;
    const bool fullTile = (row0 + 16 <= n);   // uniform per wave
    const bool inb = (m < n);

    v8f acc = {0.f, 0.f, 0.f, 0.f, 0.f, 0.f, 0.f, 0.f};
    for (int k0 = 0; k0 < 128; k0 += 4) {
        const int ka = k0 + (half << 1);
        v2f a;
        if (inb) { a.x = h1[(size_t)m * 128 + ka];
                   a.y = h1[(size_t)m * 128 + ka + 1]; }
        else     { a.x = 0.f; a.y = 0.f; }
        v2f b; b.x = W2[(size_t)ka       * 16 + mn];
               b.y = W2[(size_t)(ka + 1) * 16 + mn];
        acc = __builtin_amdgcn_wmma_f32_16x16x4_f32(
            false, a, false, b, (short)0, acc, false, false);
    }
    const int mBase = half << 3;
    if (fullTile) {
#pragma unroll
        for (int r = 0; r < 8; ++r)
            h2[(size_t)(row0 + mBase + r) * 16 + mn] = acc[r];
    } else {
#pragma unroll
        for (int r = 0; r < 8; ++r) {
            const int row = row0 + mBase + r;
            if (row < n) h2[(size_t)row * 16 + mn] = acc[r];
        }
    }
}

// ---------------------------------------------- edge scatter, layer 2 (C=16)
// Thread per (edge, 4-channel quad): float4 gather + 4 atomics.
__global__ void k_agg2(const int* __restrict__ srcIdx, const int* __restrict__ dstIdx,
                       const float* __restrict__ norm, const float* __restrict__ h2,
                       float* __restrict__ agg, int E) {
    const int t = blockIdx.x * blockDim.x + threadIdx.x;   // E*4 threads
    if (t >= E * 4) return;
    const int e = t >> 2;
    const int q = (t & 3) << 2;
    const int   s   = srcIdx[e];
    const int   d   = dstIdx[e];
    const float nrm = norm[e];
    const float4 v = *(const float4*)(h2 + (size_t)s * 16 + q);
    float* ap = agg + (size_t)d * 16 + q;
    unsafeAtomicAdd(ap + 0, v.x * nrm);
    unsafeAtomicAdd(ap + 1, v.y * nrm);
    unsafeAtomicAdd(ap + 2, v.z * nrm);
    unsafeAtomicAdd(ap + 3, v.w * nrm);
}

// ------------------------- self-loop + bias + log_softmax(16) -> out
__global__ void k_final(const float* __restrict__ h2, const float* __restrict__ agg2,
                        const float* __restrict__ dis, const float* __restrict__ b2,
                        float* __restrict__ out, int n) {
    const int i = blockIdx.x * blockDim.x + threadIdx.x;
    if (i >= n) return;
    const float d2 = dis[i] * dis[i];
    float z[16];
    float mx = -INFINITY;
#pragma unroll
    for (int c = 0; c < 16; ++c) {
        z[c] = fmaf(h2[(size_t)i * 16 + c], d2, agg2[(size_t)i * 16 + c]) + b2[c];
        mx = fmaxf(mx, z[c]);
    }
    float sum = 0.f;
#pragma unroll
    for (int c = 0; c < 16; ++c) sum += expf(z[c] - mx);
    const float lse = mx + logf(sum);
#pragma unroll
    for (int c = 0; c < 16; ++c) out[(size_t)i * 16 + c] = z[c] - lse;
}

// ---------------------------------------------------------------- launcher
extern "C" void kernel_launch(void* const* d_in, const int* in_sizes, int n_in,
                              void* d_out, int out_size, void* d_ws, size_t ws_size,
                              hipStream_t stream) {
    const float* x  = (const float*)d_in[0];
    const int*   ei = (const int*)d_in[1];   // [2,E] int32
    const float* W1 = (const float*)d_in[2];
    const float* b1 = (const float*)d_in[3];
    const float* W2 = (const float*)d_in[4];
    const float* b2 = (const float*)d_in[5];

    const int N = in_sizes[0] / 256;
    const int E = in_sizes[1] / 2;
    const int* srcIdx = ei;        // row 0
    const int* dstIdx = ei + E;    // row 1

    // carve workspace (all offsets 256B aligned)
    char*  ws  = (char*)d_ws;
    size_t off = 0;
    auto carve = [&](size_t bytes) -> float* {
        float* p = (float*)(ws + off);
        off += (bytes + 255) & ~(size_t)255;
        return p;
    };
    float* deg  = carve((size_t)N * 4);
    float* dis  = carve((size_t)N * 4);
    float* nrm  = carve((size_t)E * 4);
    float* h    = carve((size_t)N * 128 * 4);
    float* agg1 = carve((size_t)N * 128 * 4);   // becomes h1 after relu (in place)
    float* h2   = carve((size_t)N * 16 * 4);
    float* agg2 = carve((size_t)N * 16 * 4);
    (void)ws_size;

    const int T = 256;

    // degrees (self-loop folded in as init value 1.0)
    k_fill <<<(N + T - 1) / T, T, 0, stream>>>(deg, 1.0f, N);
    k_deg  <<<(E + T - 1) / T, T, 0, stream>>>(dstIdx, deg, E);
    k_rsqrt<<<(N + T - 1) / T, T, 0, stream>>>(deg, dis, N);
    k_norm <<<(E + T - 1) / T, T, 0, stream>>>(srcIdx, dstIdx, dis, nrm, E);

    // layer 1
    k_fill <<<((N * 128) + T - 1) / T, T, 0, stream>>>(agg1, 0.0f, N * 128);
    k_gemm1<<<(N + 15) / 16, T, 0, stream>>>(x, W1, h, N);
    k_agg1 <<<(E + 7) / 8, T, 0, stream>>>(srcIdx, dstIdx, nrm, h, agg1, E);
    k_bias_relu<<<((N * 32) + T - 1) / T, T, 0, stream>>>(h, dis, b1, agg1, N);

    // layer 2
    k_fill <<<((N * 16) + T - 1) / T, T, 0, stream>>>(agg2, 0.0f, N * 16);
    k_gemm2<<<(N + 127) / 128, T, 0, stream>>>(agg1, W2, h2, N);
    k_agg2 <<<((E * 4) + T - 1) / T, T, 0, stream>>>(srcIdx, dstIdx, nrm, h2, agg2, E);

    // log-softmax output
    k_final<<<(N + T - 1) / T, T, 0, stream>>>(h2, agg2, dis, b2, (float*)d_out, N);
}